// MinLSTM_47794396070432
// MI455X (gfx1250) — compile-verified
//
#include <hip/hip_runtime.h>
#include <math.h>

typedef __attribute__((ext_vector_type(2))) float v2f;
typedef __attribute__((ext_vector_type(8))) float v8f;

constexpr int kB      = 4;
constexpr int kT      = 8192;
constexpr int kD      = 512;          // D_IN == D_H == 512
constexpr int kBT     = kB * kT;      // 32768 token rows
constexpr int kNCH    = kB * kD;      // 2048 scan channels
constexpr int kNCHUNK = 32;           // chunks along T
constexpr int kCLEN   = kT / kNCHUNK; // 256 steps per chunk
constexpr int kPad    = 17;           // LDS column stride (gcd(17,64)=1)

// ---------- log-space gate helpers (match the JAX reference) ----------
__device__ __forceinline__ float softplusf(float z) {
    // log(1+exp(z)) stable form
    return fmaxf(z, 0.0f) + log1pf(expf(-fabsf(z)));
}
__device__ __forceinline__ float log_gf(float x) {
    // log g(x), g(x) = x>=0 ? x+0.5 : sigmoid(x)
    return (x >= 0.0f) ? logf(x + 0.5f) : -softplusf(-x);
}
__device__ __forceinline__ float laddexp(float x, float y) {
    float m = fmaxf(x, y);
    if (m == -INFINITY) return -INFINITY;
    return m + log1pf(expf(-fabsf(x - y)));
}

// ---------------------------------------------------------------------
// Kernel 1: fused 3-gate GEMM (f32 WMMA) + log-space gate epilogue.
// Block = 256 threads (8 waves). Block owns 16 token rows; wave w owns
// output columns [w*64, w*64+64) for all three gates -> 12 v8f
// accumulators = 96 VGPRs/lane.  ks loop kept rolled so B-fragment
// pipelining stays bounded.  Epilogue drains accumulators through LDS
// (2 phases x 52KB) so their live ranges end at a ds_store instead of
// forcing the register allocator to spill 96 VGPRs to scratch.
// Emits LF = log_f and LV = log_i + log_g(h_pre), both (BT, D) f32.
// ---------------------------------------------------------------------
__global__ __launch_bounds__(256, 1) void minlstm_gates_gemm(
    const float* __restrict__ x,
    const float* __restrict__ Wf, const float* __restrict__ bf,
    const float* __restrict__ Wi, const float* __restrict__ bi,
    const float* __restrict__ Wh, const float* __restrict__ bh,
    float* __restrict__ LF, float* __restrict__ LV)
{
    __shared__ float ldsA[16 * 32];            // 16 rows x Kc=32 A staging
    __shared__ float ldsP[3 * 256 * kPad];     // pre-activation drain (per phase)

    const int lane  = threadIdx.x & 31;
    const int wave  = threadIdx.x >> 5;
    const int nbase = wave * 64;
    const int mrow0 = blockIdx.x * 16;

    const float* Wg[3] = { Wf, Wi, Wh };

    v8f acc[3][4];
    #pragma unroll
    for (int g = 0; g < 3; ++g)
        #pragma unroll
        for (int t = 0; t < 4; ++t)
            acc[g][t] = v8f{};

    // f32 16x16x4 WMMA operand layout (wave32):
    //   lanes 0-15 : row/col = lane,    K = {k, k+1}
    //   lanes 16-31: row/col = lane-16, K = {k+2, k+3}
    const int arow  = lane & 15;
    const int khalf = (lane >> 4) * 2;

    #pragma unroll 1
    for (int k0 = 0; k0 < kD; k0 += 32) {
        __syncthreads();
        // cooperative stage of 16x32 A tile (512 floats, 2 per thread)
        #pragma unroll
        for (int e = 0; e < 2; ++e) {
            int idx = threadIdx.x + e * 256;
            int r = idx >> 5, c = idx & 31;
            ldsA[idx] = x[(size_t)(mrow0 + r) * kD + k0 + c];
        }
        // prefetch next A tile slice into the cache hierarchy
        if (k0 + 32 < kD)
            __builtin_prefetch(&x[(size_t)(mrow0 + (threadIdx.x >> 5)) * kD + k0 + 32], 0, 1);
        __syncthreads();

        #pragma unroll 1
        for (int ks = 0; ks < 8; ++ks) {
            const int kk = ks * 4 + khalf;   // even -> 8B aligned v2f reads
            const v2f a = *(const v2f*)&ldsA[arow * 32 + kk];

            // gather all 12 B fragments first (one load clause), then issue
            // 12 independent back-to-back WMMAs (same A, disjoint D).
            v2f bfr[12];
            #pragma unroll
            for (int g = 0; g < 3; ++g) {
                const float* __restrict__ W = Wg[g];
                #pragma unroll
                for (int t = 0; t < 4; ++t) {
                    const int ncol = nbase + t * 16 + arow;
                    bfr[g * 4 + t] = *(const v2f*)&W[(size_t)ncol * kD + k0 + kk];
                }
            }
            #pragma unroll
            for (int g = 0; g < 3; ++g)
                #pragma unroll
                for (int t = 0; t < 4; ++t)
                    acc[g][t] = __builtin_amdgcn_wmma_f32_16x16x4_f32(
                        false, a, false, bfr[g * 4 + t], (short)0, acc[g][t],
                        false, false);
        }
    }

    // ------------------------------------------------------------------
    // Epilogue, 2 phases of 2 N-tiles. C/D layout: VGPR r holds
    // M = rbase + r with rbase = (lane>>4)*8, N = nbase + 16*t + (lane&15).
    // Phase p drains tiles t = 2p, 2p+1 into LDS (column-major, stride 17,
    // conflict-free), then all threads recompute gates from LDS with
    // minimal register pressure and store LF/LV coalesced.
    // ------------------------------------------------------------------
    const int rbase = (lane >> 4) * 8;
    #pragma unroll
    for (int p = 0; p < 2; ++p) {
        __syncthreads();
        #pragma unroll
        for (int g = 0; g < 3; ++g) {
            #pragma unroll
            for (int th = 0; th < 2; ++th) {
                const int lcol = wave * 32 + th * 16 + arow;   // 0..255
                float* bp = &ldsP[(g * 256 + lcol) * kPad + rbase];
                const v8f v = acc[g][p * 2 + th];
                #pragma unroll
                for (int r = 0; r < 8; ++r) bp[r] = v[r];
            }
        }
        __syncthreads();

        #pragma unroll 1
        for (int i = 0; i < 16; ++i) {           // 4096 elements / 256 threads
            const int idx  = i * 256 + (int)threadIdx.x;   // 0..4095
            const int row  = idx >> 8;                      // 0..15
            const int lcol = idx & 255;
            const int col  = (lcol >> 5) * 64 + p * 32 + (lcol & 31);
            const float fp = ldsP[(0 * 256 + lcol) * kPad + row] + bf[col];
            const float ip = ldsP[(1 * 256 + lcol) * kPad + row] + bi[col];
            const float hp = ldsP[(2 * 256 + lcol) * kPad + row] + bh[col];
            const float diff = softplusf(-fp) - softplusf(-ip);
            const float lf   = -softplusf(diff);      // log forget gate
            const float li   = -softplusf(-diff);     // log input gate
            const float lv   = li + log_gf(hp);       // log value term
            const size_t o = (size_t)(mrow0 + row) * kD + col;
            LF[o] = lf;
            LV[o] = lv;
        }
    }
}

// ---------------------------------------------------------------------
// Chunked parallel scan over T.  Recurrence: L_t = laddexp(lf_t + L, lv_t).
// Chunk summary (A = sum lf, B = chunk scan from -inf) composes as
// state' = laddexp(A + state, B)  (associative).
// ---------------------------------------------------------------------
__global__ __launch_bounds__(256) void minlstm_scan_pass1(
    const float* __restrict__ LF, const float* __restrict__ LV,
    float* __restrict__ Acar, float* __restrict__ Bcar)
{
    const int ch = blockIdx.x * 256 + threadIdx.x;   // 0..2047
    const int j  = blockIdx.y;                       // chunk
    const int b  = ch >> 9;
    const int h  = ch & (kD - 1);
    size_t base = ((size_t)b * kT + (size_t)j * kCLEN) * kD + h;

    float A = 0.0f;
    float L = -INFINITY;
    for (int t = 0; t < kCLEN; ++t) {
        const size_t o = base + (size_t)t * kD;      // coalesced across lanes
        const float lf = LF[o];
        const float lv = LV[o];
        A += lf;
        L = laddexp(lf + L, lv);
    }
    Acar[ch * kNCHUNK + j] = A;
    Bcar[ch * kNCHUNK + j] = L;
}

__global__ __launch_bounds__(256) void minlstm_scan_pass2(
    const float* __restrict__ h0,
    const float* __restrict__ Acar, const float* __restrict__ Bcar,
    float* __restrict__ Pref)
{
    const int ch = blockIdx.x * 256 + threadIdx.x;   // 0..2047
    float P = log_gf(h0[ch]);                        // log_g(h_0), state before t=1
    #pragma unroll
    for (int j = 0; j < kNCHUNK; ++j) {
        Pref[ch * kNCHUNK + j] = P;                  // incoming state for chunk j
        P = laddexp(Acar[ch * kNCHUNK + j] + P, Bcar[ch * kNCHUNK + j]);
    }
}

__global__ __launch_bounds__(256) void minlstm_scan_pass3(
    const float* __restrict__ LF, const float* __restrict__ LV,
    const float* __restrict__ Pref, float* __restrict__ out)
{
    const int ch = blockIdx.x * 256 + threadIdx.x;
    const int j  = blockIdx.y;
    const int b  = ch >> 9;
    const int h  = ch & (kD - 1);
    size_t base = ((size_t)b * kT + (size_t)j * kCLEN) * kD + h;

    float L = Pref[ch * kNCHUNK + j];
    for (int t = 0; t < kCLEN; ++t) {
        const size_t o = base + (size_t)t * kD;
        L = laddexp(LF[o] + L, LV[o]);
        out[o] = expf(L);                            // h_t, t = 1..T
    }
}

// ---------------------------------------------------------------------
extern "C" void kernel_launch(void* const* d_in, const int* in_sizes, int n_in,
                              void* d_out, int out_size, void* d_ws, size_t ws_size,
                              hipStream_t stream) {
    const float* x  = (const float*)d_in[0];
    const float* h0 = (const float*)d_in[1];
    const float* Wf = (const float*)d_in[2];
    const float* bf = (const float*)d_in[3];
    const float* Wi = (const float*)d_in[4];
    const float* bi = (const float*)d_in[5];
    const float* Wh = (const float*)d_in[6];
    const float* bh = (const float*)d_in[7];
    float* out = (float*)d_out;

    float* ws   = (float*)d_ws;
    float* LF   = ws;                                   // 16.7M floats
    float* LV   = LF + (size_t)kBT * kD;                // 16.7M floats
    float* Acar = LV + (size_t)kBT * kD;                // 2048*32
    float* Bcar = Acar + (size_t)kNCH * kNCHUNK;
    float* Pref = Bcar + (size_t)kNCH * kNCHUNK;

    minlstm_gates_gemm<<<kBT / 16, 256, 0, stream>>>(x, Wf, bf, Wi, bi, Wh, bh,
                                                     LF, LV);
    dim3 gscan(kNCH / 256, kNCHUNK);
    minlstm_scan_pass1<<<gscan, 256, 0, stream>>>(LF, LV, Acar, Bcar);
    minlstm_scan_pass2<<<kNCH / 256, 256, 0, stream>>>(h0, Acar, Bcar, Pref);
    minlstm_scan_pass3<<<gscan, 256, 0, stream>>>(LF, LV, Pref, out);
}